// CurvatureMap_83537113907616
// MI455X (gfx1250) — compile-verified
//
#include <hip/hip_runtime.h>

typedef __attribute__((ext_vector_type(2))) float v2f;
typedef __attribute__((ext_vector_type(4))) float v4f;
typedef __attribute__((ext_vector_type(8))) float v8f;

#define NB 8
#define NC 64
#define NH 512
#define NW 512
#define NHW (NH * NW)
#define CEPS 1e-8f

// ---------------------------------------------------------------------------
// Kernel 1: channel mean over 64 channels. Pure bandwidth (512 MiB stream).
// Each thread owns 4 consecutive pixels -> float4 loads, NT hint so the big
// input stream does not evict the 8 MiB mean we want resident in L2.
// ---------------------------------------------------------------------------
__global__ void mean_kernel(const float* __restrict__ x, float* __restrict__ mean) {
    long t = (long)blockIdx.x * blockDim.x + threadIdx.x;   // 524288 threads
    long p = t * 4;                                         // flat pixel index (b*NHW + pix)
    int  b   = (int)(p / NHW);
    int  pix = (int)(p % NHW);
    const float* base = x + ((long)b * NC) * NHW + pix;
    v4f acc = {0.f, 0.f, 0.f, 0.f};
#pragma unroll 8
    for (int c = 0; c < NC; ++c) {
        v4f v = __builtin_nontemporal_load((const v4f*)(base + (long)c * NHW));
        acc += v;
    }
    acc *= (1.0f / (float)NC);
    *(v4f*)(mean + p) = acc;
}

// ---------------------------------------------------------------------------
// Kernel 2: 3x3 x 5-filter conv as WMMA f32 16x16x4 (K = 9 padded to 12,
// M = 5 filters padded to 16, N = 16 pixels), then pointwise curvature.
// Block = 256 threads = 8 waves; block tile = 8 rows x 512 cols of one image.
// ---------------------------------------------------------------------------
__global__ void curv_kernel(const float* __restrict__ mean,
                            const float* __restrict__ alpha_p,
                            const float* __restrict__ beta_p,
                            float* __restrict__ out) {
    const int PITCH = NW + 2;                 // 514
    __shared__ float lds[10 * (NW + 2)];      // (8+2) x 514 halo tile, 20.5 KB

    const int img   = blockIdx.y;             // 0..7
    const int strip = blockIdx.x;             // 0..63
    const int y0    = strip * 8;
    const int tid   = threadIdx.x;

    const float* m = mean + (long)img * NHW;

    // Cooperative edge-clamped halo load.
    for (int i = tid; i < 10 * PITCH; i += 256) {
        int ty = i / PITCH, tx = i % PITCH;
        int sy = min(max(y0 + ty - 1, 0), NH - 1);
        int sx = min(max(tx - 1, 0), NW - 1);
        lds[i] = m[sy * NW + sx];
    }
    __syncthreads();

    const int wave = tid >> 5;                // 0..7 : row within strip
    const int lane = tid & 31;
    const int half = lane >> 4;               // ISA: lanes 16..31 carry K+2
    const int mrow = lane & 15;               // A's M row == B/D's N column

    // Sobel-family taps (row-major 3x3), rows of A: [kx, ky, kxx, kxy, kyy].
    // kx = s^T d, ky = d^T s, kxx = s^T d2, kxy = d^T d, kyy = d2^T s
    // with d = [-1,0,1]/2, s = [1,2,1]/4, d2 = [1,-2,1].
    const float W0[9] = {-0.125f, 0.f, 0.125f, -0.25f, 0.f, 0.25f, -0.125f, 0.f, 0.125f}; // I_x
    const float W1[9] = {-0.125f, -0.25f, -0.125f, 0.f, 0.f, 0.f, 0.125f, 0.25f, 0.125f}; // I_y
    const float W2[9] = {0.25f, -0.5f, 0.25f, 0.5f, -1.f, 0.5f, 0.25f, -0.5f, 0.25f};     // I_xx
    const float W3[9] = {0.25f, 0.f, -0.25f, 0.f, 0.f, 0.f, -0.25f, 0.f, 0.25f};          // I_xy
    const float W4[9] = {0.25f, 0.5f, 0.25f, -0.5f, -1.f, -0.5f, 0.25f, 0.5f, 0.25f};     // I_yy

    // Precompute the three A fragments (K chunks 0..3, 4..7, 8..11).
    // A layout: lane holds M = lane&15; VGPR0 -> K = 4c + 2*half, VGPR1 -> +1.
    v2f afrag[3];
#pragma unroll
    for (int c = 0; c < 3; ++c) {
        int k0 = 4 * c + 2 * half;
        float ax = 0.f, ay = 0.f;
        if (mrow < 5) {
            const float* Wm = (mrow == 0) ? W0 : (mrow == 1) ? W1 : (mrow == 2) ? W2
                            : (mrow == 3) ? W3 : W4;
            if (k0 < 9)     ax = Wm[k0];
            if (k0 + 1 < 9) ay = Wm[k0 + 1];
        }
        afrag[c] = (v2f){ax, ay};
    }

    const float alpha = *alpha_p;
    const float beta  = *beta_p;

    float* outK = out;
    float* outH = out + (long)NB * NHW;
    float* outk = out + 2L * (long)NB * NHW;
    const long obase = (long)img * NHW + (long)(y0 + wave) * NW;

    // B layout mirrors A: lane holds N = lane&15; VGPR0 -> K = 4c + 2*half.
    // im2col row k -> tap (i,j) = (k/3, k%3); lds value = lds[(wave+i)*PITCH + x0+n+j].
    const int lb = wave * PITCH + mrow;       // base LDS offset for this lane

    for (int g = 0; g < NW / 16; ++g) {
        const int x0 = g * 16;
        v8f d = {0.f, 0.f, 0.f, 0.f, 0.f, 0.f, 0.f, 0.f};

        // chunk 0: K = {0,1} (half 0) / {2,3} (half 1): taps (0,0),(0,1) / (0,2),(1,0)
        {
            float bx = lds[lb + x0 + (half ? 2 : 0)];
            float by = lds[lb + x0 + (half ? PITCH + 0 : 1)];
            v2f bfrag = {bx, by};
            d = __builtin_amdgcn_wmma_f32_16x16x4_f32(false, afrag[0], false, bfrag,
                                                      (short)0, d, false, false);
        }
        // chunk 1: K = {4,5} / {6,7}: taps (1,1),(1,2) / (2,0),(2,1)
        {
            float bx = lds[lb + x0 + (half ? 2 * PITCH + 0 : PITCH + 1)];
            float by = lds[lb + x0 + (half ? 2 * PITCH + 1 : PITCH + 2)];
            v2f bfrag = {bx, by};
            d = __builtin_amdgcn_wmma_f32_16x16x4_f32(false, afrag[1], false, bfrag,
                                                      (short)0, d, false, false);
        }
        // chunk 2: K = {8,9} / {10,11}: tap (2,2) valid only for half 0, rest zero-pad
        {
            float bx = half ? 0.f : lds[lb + x0 + 2 * PITCH + 2];
            v2f bfrag = {bx, 0.f};
            d = __builtin_amdgcn_wmma_f32_16x16x4_f32(false, afrag[2], false, bfrag,
                                                      (short)0, d, false, false);
        }

        // D layout: lanes 0..15 hold M = 0..7 in VGPRs 0..7 for column N = lane.
        // So lane n owns I_x..I_yy for pixel x0+n in d[0..4].
        if (lane < 16) {
            float Ix = d[0], Iy = d[1], Ixx = d[2], Ixy = d[3], Iyy = d[4];
            float Ix2 = Ix * Ix, Iy2 = Iy * Iy;
            float g1 = 1.f + Ix2 + Iy2;
            float Kv = (Ixx * Iyy - Ixy * Ixy) / (g1 * g1 + CEPS);
            float Hv = ((1.f + Ix2) * Iyy - 2.f * Ix * Iy * Ixy + (1.f + Iy2) * Ixx)
                       / (2.f * g1 * __builtin_sqrtf(g1) + CEPS);
            float kap = alpha * __builtin_fabsf(Kv) + beta * __builtin_fabsf(Hv);
            long o = obase + x0 + mrow;
            outK[o] = Kv;
            outH[o] = Hv;
            outk[o] = kap;
        }
    }
}

// ---------------------------------------------------------------------------
extern "C" void kernel_launch(void* const* d_in, const int* in_sizes, int n_in,
                              void* d_out, int out_size, void* d_ws, size_t ws_size,
                              hipStream_t stream) {
    const float* x       = (const float*)d_in[0];
    const float* alpha_p = (const float*)d_in[1];
    const float* beta_p  = (const float*)d_in[2];
    float* out  = (float*)d_out;
    float* mean = (float*)d_ws;                 // needs NB*NHW*4 = 8 MiB scratch

    // Kernel 1: 8*512*512/4 lanes = 524288 threads.
    mean_kernel<<<2048, 256, 0, stream>>>(x, mean);

    // Kernel 2: 64 strips x 8 images, 256 threads (8 waves) each.
    dim3 grid(NH / 8, NB);
    curv_kernel<<<grid, 256, 0, stream>>>(mean, alpha_p, beta_p, out);
}